// MaxSim_59734405153137
// MI455X (gfx1250) — compile-verified
//
#include <hip/hip_runtime.h>
#include <hip/hip_bf16.h>

#define BATCH   32
#define QLEN    32
#define HID     128
#define DOCLEN  128
#define KCAND   256

#define SLABW   32          // h-columns per slab
#define NSLAB   (HID / SLABW)
#define DSTRIDE 36          // padded slab row stride (floats): 16B-aligned, conflict-free
#define QSTRIDE 132         // padded Q row stride (floats)

typedef __attribute__((ext_vector_type(2))) float v2f;
typedef __attribute__((ext_vector_type(8))) float v8f;

typedef int v4i __attribute__((vector_size(16)));
typedef v4i __attribute__((address_space(1)))* gv4i_p;   // global int4*
typedef v4i __attribute__((address_space(3)))* lv4i_p;   // LDS int4*

#define NEG_INF __uint_as_float(0xff800000u)

#if defined(__AMDGCN__) && __has_builtin(__builtin_amdgcn_global_load_async_to_lds_b128)
#define USE_ASYNC_LDS 1
#else
#define USE_ASYNC_LDS 0
#endif

__device__ __forceinline__ void async_copy_b128(const float* g, float* l) {
#if USE_ASYNC_LDS
    __builtin_amdgcn_global_load_async_to_lds_b128(
        (gv4i_p)(const_cast<float*>(g)), (lv4i_p)(l), 0, 0);
#else
    (void)g; (void)l;
#endif
}

__device__ __forceinline__ void wait_async_le4() {
#if __has_builtin(__builtin_amdgcn_s_wait_asynccnt)
    __builtin_amdgcn_s_wait_asynccnt(4);
#else
    asm volatile("s_wait_asynccnt 0x4" ::: "memory");
#endif
}

__device__ __forceinline__ void wait_async_le0() {
#if __has_builtin(__builtin_amdgcn_s_wait_asynccnt)
    __builtin_amdgcn_s_wait_asynccnt(0);
#else
    asm volatile("s_wait_asynccnt 0x0" ::: "memory");
#endif
}

// ---------------- Phase 1: pids = emb2pid[topk]; per-row sort, dups -> -1 ---
__global__ void maxsim_pid_sort_kernel(const int* __restrict__ topk,
                                       const int* __restrict__ emb2pid,
                                       int* __restrict__ ws_pids) {
    __shared__ int s[KCAND];
    const int b = blockIdx.x;
    const int t = threadIdx.x;
    s[t] = emb2pid[topk[b * KCAND + t]];
    __syncthreads();
    for (int ksz = 2; ksz <= KCAND; ksz <<= 1) {
        for (int j = ksz >> 1; j > 0; j >>= 1) {
            int ixj = t ^ j;
            if (ixj > t) {
                int a = s[t], c = s[ixj];
                bool asc = ((t & ksz) == 0);
                if ((a > c) == asc) { s[t] = c; s[ixj] = a; }
            }
            __syncthreads();
        }
    }
    int v = s[t];
    bool dup = (t > 0) && (s[t - 1] == v);
    ws_pids[b * KCAND + t] = dup ? -1 : v;
}

// ---------------- Phase 2 helpers ------------------------------------------
__device__ __forceinline__ void issue_slab_async(const float* Dg, float* sD,
                                                 int c0, int t) {
    // 128 rows x 32 cols slab: 4 x b128 async copies per thread
    #pragma unroll
    for (int i = 0; i < 4; ++i) {
        int f   = 4 * (t + i * 256);     // over 4096 floats
        int row = f >> 5;                // /32
        int col = f & 31;
        async_copy_b128(&Dg[(size_t)row * HID + c0 + col], &sD[row * DSTRIDE + col]);
    }
}

__device__ __forceinline__ void stage_slab_sync(const float* Dg, float* sD,
                                                int c0, int t) {
    float4 tmp[4];
    #pragma unroll
    for (int i = 0; i < 4; ++i) {
        int f   = 4 * (t + i * 256);
        int row = f >> 5;
        int col = f & 31;
        tmp[i] = *(const float4*)&Dg[(size_t)row * HID + c0 + col];
    }
    #pragma unroll
    for (int i = 0; i < 4; ++i) {
        int f   = 4 * (t + i * 256);
        int row = f >> 5;
        int col = f & 31;
        *(float4*)&sD[row * DSTRIDE + col] = tmp[i];
    }
}

__device__ __forceinline__ void compute_slab(const float* sD, const float* sQ,
                                             int c0, int l16, int half, int d0,
                                             v8f& cl, v8f& ch) {
    const float* bptr = &sD[(d0 + l16) * DSTRIDE + 2 * half];
    const float* alop = &sQ[l16 * QSTRIDE + c0 + 2 * half];
    const float* ahip = &sQ[(16 + l16) * QSTRIDE + c0 + 2 * half];
    #pragma unroll
    for (int k0 = 0; k0 < SLABW; k0 += 4) {
        v2f bf = *(const v2f*)(bptr + k0);   // B[k][n] = D[n][h]
        v2f al = *(const v2f*)(alop + k0);   // A[m][k] = Q[q][h]
        v2f ah = *(const v2f*)(ahip + k0);
        cl = __builtin_amdgcn_wmma_f32_16x16x4_f32(
                 false, al, false, bf, (short)0, cl, false, false);
        ch = __builtin_amdgcn_wmma_f32_16x16x4_f32(
                 false, ah, false, bf, (short)0, ch, false, false);
    }
}

// ---------------- Phase 2: per-(b,candidate) MaxSim score via f32 WMMA ------
__global__ void maxsim_score_kernel(const float* __restrict__ qv,
                                    const float* __restrict__ vectors,
                                    const int* __restrict__ ws_pids,
                                    float* __restrict__ ws_scores) {
    const int kc = blockIdx.x;
    const int b  = blockIdx.y;
    const int t  = threadIdx.x;
    const int pid = ws_pids[b * KCAND + kc];

    if (pid < 0) {                       // duplicate slot: masked to -inf
        if (t == 0) ws_scores[b * KCAND + kc] = NEG_INF;
        return;
    }

    extern __shared__ float smem[];
    float* sD0  = smem;                          // 128 x 36
    float* sD1  = sD0 + DOCLEN * DSTRIDE;        // 128 x 36
    float* sQ   = sD1 + DOCLEN * DSTRIDE;        // 32 x 132
    float* smax = sQ + QLEN * QSTRIDE;           // 32 q x 8 waves
    float* sDbuf[2] = { sD0, sD1 };

    const float* Dg = vectors + (size_t)pid * (DOCLEN * HID);
    const float* Qg = qv + (size_t)b * (QLEN * HID);

    const int w    = t >> 5;
    const int lane = t & 31;
    const int half = lane >> 4;          // A/B: lanes 16-31 hold K+2,K+3
    const int l16  = lane & 15;
    const int d0   = w * 16;

    v8f cl = {};                         // q rows 0..15
    v8f ch = {};                         // q rows 16..31

#if USE_ASYNC_LDS
    // Issue Q (4 copies), slab0, slab1 -> ASYNCcnt = 12 per thread.
    #pragma unroll
    for (int i = 0; i < 4; ++i) {
        int f   = 4 * (t + i * 256);
        int row = f >> 7;
        int col = f & 127;
        async_copy_b128(&Qg[f], &sQ[row * QSTRIDE + col]);
    }
    issue_slab_async(Dg, sDbuf[0], 0 * SLABW, t);
    issue_slab_async(Dg, sDbuf[1], 1 * SLABW, t);

    #pragma unroll
    for (int s = 0; s < NSLAB; ++s) {
        if (s + 1 < NSLAB) wait_async_le4();   // retires (Q+)slab s, in order
        else               wait_async_le0();
        __syncthreads();                       // all waves' copies visible
        compute_slab(sDbuf[s & 1], sQ, s * SLABW, l16, half, d0, cl, ch);
        __syncthreads();                       // all waves done reading buffer
        if (s + 2 < NSLAB)
            issue_slab_async(Dg, sDbuf[s & 1], (s + 2) * SLABW, t);
    }
#else
    // Fallback: VGPR-staged copies (all loads issued before stores).
    {
        float4 qtmp[4];
        #pragma unroll
        for (int i = 0; i < 4; ++i) {
            int f = 4 * (t + i * 256);
            qtmp[i] = *(const float4*)&Qg[f];
        }
        #pragma unroll
        for (int i = 0; i < 4; ++i) {
            int f   = 4 * (t + i * 256);
            int row = f >> 7;
            int col = f & 127;
            *(float4*)&sQ[row * QSTRIDE + col] = qtmp[i];
        }
    }
    #pragma unroll
    for (int s = 0; s < NSLAB; ++s) {
        __syncthreads();
        stage_slab_sync(Dg, sDbuf[0], s * SLABW, t);
        __syncthreads();
        compute_slab(sDbuf[0], sQ, s * SLABW, l16, half, d0, cl, ch);
    }
#endif

    // Max over this wave's 16 d-columns (intra-half reduction), per q row.
    #pragma unroll
    for (int r = 0; r < 8; ++r) {
        float v = cl[r];
        v = fmaxf(v, __shfl_xor(v, 1, 32));
        v = fmaxf(v, __shfl_xor(v, 2, 32));
        v = fmaxf(v, __shfl_xor(v, 4, 32));
        v = fmaxf(v, __shfl_xor(v, 8, 32));
        if (l16 == 0) smax[(r + 8 * half) * 8 + w] = v;
        float u = ch[r];
        u = fmaxf(u, __shfl_xor(u, 1, 32));
        u = fmaxf(u, __shfl_xor(u, 2, 32));
        u = fmaxf(u, __shfl_xor(u, 4, 32));
        u = fmaxf(u, __shfl_xor(u, 8, 32));
        if (l16 == 0) smax[(16 + r + 8 * half) * 8 + w] = u;
    }
    __syncthreads();

    if (t < 32) {                        // wave 0: max over waves, mean over q
        float m = smax[t * 8 + 0];
        #pragma unroll
        for (int j = 1; j < 8; ++j) m = fmaxf(m, smax[t * 8 + j]);
        m += __shfl_xor(m, 16, 32);
        m += __shfl_xor(m, 8, 32);
        m += __shfl_xor(m, 4, 32);
        m += __shfl_xor(m, 2, 32);
        m += __shfl_xor(m, 1, 32);
        if (t == 0) ws_scores[b * KCAND + kc] = m * (1.0f / QLEN);
    }
}

// ---------------- Phase 3: per-batch top-k (score desc, index asc) ----------
__global__ void maxsim_topk_kernel(const int* __restrict__ ws_pids,
                                   const float* __restrict__ ws_scores,
                                   float* __restrict__ out, int kk) {
    __shared__ float ss[KCAND];
    __shared__ int   si[KCAND];
    const int b = blockIdx.x;
    const int t = threadIdx.x;
    int   pid = ws_pids[b * KCAND + t];
    float sc  = ws_scores[b * KCAND + t];
    if (pid < 0) sc = NEG_INF;
    ss[t] = sc;
    si[t] = t;
    __syncthreads();
    for (int ksz = 2; ksz <= KCAND; ksz <<= 1) {
        for (int j = ksz >> 1; j > 0; j >>= 1) {
            int ixj = t ^ j;
            if (ixj > t) {
                float s0 = ss[t], s1 = ss[ixj];
                int   i0 = si[t], i1 = si[ixj];
                bool dir = ((t & ksz) == 0);
                bool b10 = (s1 > s0) || (s1 == s0 && i1 < i0);
                bool b01 = (s0 > s1) || (s0 == s1 && i0 < i1);
                if (dir ? b10 : b01) {
                    ss[t] = s1; ss[ixj] = s0;
                    si[t] = i1; si[ixj] = i0;
                }
            }
            __syncthreads();
        }
    }
    if (t < kk) {
        int idx = si[t];
        out[b * kk + t] = (float)ws_pids[b * KCAND + idx];  // pids block
        out[BATCH * kk + b * kk + t] = ss[t];               // scores block
    }
}

extern "C" void kernel_launch(void* const* d_in, const int* in_sizes, int n_in,
                              void* d_out, int out_size, void* d_ws, size_t ws_size,
                              hipStream_t stream) {
    const float* qv      = (const float*)d_in[0];   // [32,32,128] f32
    const int*   topk    = (const int*)d_in[1];     // [32,256]
    const float* vectors = (const float*)d_in[3];   // [20000,128,128] f32
    const int*   emb2pid = (const int*)d_in[4];     // [2560000]

    float* out = (float*)d_out;
    int*   ws_pids   = (int*)d_ws;                            // 32*256 ints
    float* ws_scores = (float*)((char*)d_ws + BATCH * KCAND * sizeof(int));

    int kk = out_size / (2 * BATCH);                // k recovered from output size

    maxsim_pid_sort_kernel<<<BATCH, KCAND, 0, stream>>>(topk, emb2pid, ws_pids);

    size_t shmem = (size_t)(2 * DOCLEN * DSTRIDE + QLEN * QSTRIDE + 32 * 8)
                   * sizeof(float);                 // 54,784 bytes
    maxsim_score_kernel<<<dim3(KCAND, BATCH), 256, shmem, stream>>>(
        qv, vectors, ws_pids, ws_scores);

    maxsim_topk_kernel<<<BATCH, KCAND, 0, stream>>>(ws_pids, ws_scores, out, kk);
}